// Memory_18227841204789
// MI455X (gfx1250) — compile-verified
//
#include <hip/hip_runtime.h>
#include <math.h>

// ---------------------------------------------------------------------------
// Memory module (MI455X / gfx1250).
//   text_token  [32768,1024] f32   d_in[0]
//   image_token [32768,1024] f32   d_in[1]
//   cache       [25,1024]    f32   d_in[2]
//   W_ext       [1024,2048]  f32   d_in[3]
// Outputs (flat in d_out, f32):
//   text_fine_feature [32768,1024], loss [1], new_cache [25,1024]
//
// Algebra used:
//   fine@W2^T = (p@cache)@W2^T = p@(cache@W2^T) = p@G   (G precomputed, 25x1024)
//   => out = ALPHA*(text@W1^T + p@G) + text : one WMMA GEMM with K = 1024+25(pad 32)
//   softmax(s,axis=0)[i,a]/colmax[a] = exp(s[i,a] - max_i s[i,a])  (no normalizer)
//
// Workspace (~16.4 MB assumed available in d_ws).
// ---------------------------------------------------------------------------

typedef __attribute__((ext_vector_type(2))) float v2f;
typedef __attribute__((ext_vector_type(8))) float v8f;

#define C_ROWS 32768
#define DIM    1024
#define MSLOTS 25
#define ALPHA_ 0.2f
#define MOM_   0.8f

// ---- workspace layout (units of 4 bytes) ----
#define OFF_P      0u                                  // [C,25] softmax weights
#define OFF_G      (OFF_P      + C_ROWS * MSLOTS)      // [25,1024] cache@W2^T
#define OFF_ASSIGN (OFF_G      + MSLOTS * DIM)         // [C] int
#define OFF_SMAX   (OFF_ASSIGN + C_ROWS)               // [C] f32
#define OFF_INVN   (OFF_SMAX   + C_ROWS)               // [C] f32
#define OFF_CMAX   (OFF_INVN   + C_ROWS)               // [32] uint keys
#define OFF_CNT    (OFF_CMAX   + 32)                   // [32] int
#define OFF_SLOT   (OFF_CNT    + 32)                   // [25,1024] f32
#define OFF_RN     (OFF_SLOT   + MSLOTS * DIM)         // [C] f32 row sumsq -> inv norm
#define OFF_LOSS   (OFF_RN     + C_ROWS)               // [16] f32
#define OFF_PART   (OFF_LOSS   + 16)                   // [128][25*1024] partial slot sums

__device__ __forceinline__ unsigned fkey(float f) {
  unsigned u = __float_as_uint(f);
  return (u & 0x80000000u) ? ~u : (u | 0x80000000u);   // order-preserving map
}
__device__ __forceinline__ float fdec(unsigned u) {
  return __uint_as_float((u & 0x80000000u) ? (u & 0x7fffffffu) : ~u);
}

// ---------------- init: zero accumulators ----------------
__global__ void k_init(float* ws) {
  int idx = blockIdx.x * 256 + threadIdx.x;
  if (idx < C_ROWS) ws[OFF_RN + idx] = 0.0f;
  if (idx < 32) {
    ((unsigned*)(ws + OFF_CMAX))[idx] = 0u;   // key of smallest float
    ((int*)(ws + OFF_CNT))[idx] = 0;
  }
  if (idx < 16) ws[OFF_LOSS + idx] = 0.0f;
}

// ---------------- G = cache @ W2^T  (25 x 1024) ----------------
__global__ void k_G(const float* __restrict__ cache, const float* __restrict__ W,
                    float* __restrict__ ws) {
  int idx = blockIdx.x * 256 + threadIdx.x;      // 25600 threads
  int j = idx >> 10, n = idx & (DIM - 1);
  const float4* c4 = reinterpret_cast<const float4*>(cache) + j * (DIM / 4);
  const float4* w4 = reinterpret_cast<const float4*>(W) + n * 512 + 256; // W[n,1024+...]
  float s = 0.0f;
  for (int k = 0; k < DIM / 4; ++k) {
    float4 a = c4[k], b = w4[k];
    s += a.x * b.x + a.y * b.y + a.z * b.z + a.w * b.w;
  }
  ws[OFF_G + idx] = s;
}

// ---------------- p = softmax(text @ cache^T, axis=1)  (wave per row) ----------------
__global__ void k_softmax_p(const float* __restrict__ text, const float* __restrict__ cache,
                            float* __restrict__ ws) {
  const int lane = threadIdx.x & 31;
  const int row = blockIdx.x * 8 + (threadIdx.x >> 5);
  const float4* t4 = reinterpret_cast<const float4*>(text) + row * (DIM / 4);
  const float4* c4 = reinterpret_cast<const float4*>(cache);
  float part[MSLOTS];
#pragma unroll
  for (int j = 0; j < MSLOTS; ++j) part[j] = 0.0f;
  for (int k = lane; k < DIM / 4; k += 32) {
    float4 tv = t4[k];
#pragma unroll
    for (int j = 0; j < MSLOTS; ++j) {
      float4 cv = c4[j * (DIM / 4) + k];
      part[j] += tv.x * cv.x + tv.y * cv.y + tv.z * cv.z + tv.w * cv.w;
    }
  }
#pragma unroll
  for (int off = 16; off > 0; off >>= 1) {
#pragma unroll
    for (int j = 0; j < MSLOTS; ++j) part[j] += __shfl_xor(part[j], off, 32);
  }
  float m = part[0];
#pragma unroll
  for (int j = 1; j < MSLOTS; ++j) m = fmaxf(m, part[j]);
  float e[MSLOTS], sum = 0.0f;
#pragma unroll
  for (int j = 0; j < MSLOTS; ++j) { e[j] = expf(part[j] - m); sum += e[j]; }
  float inv = 1.0f / sum;
  if (lane < MSLOTS) {
    float val = 0.0f;
#pragma unroll
    for (int j = 0; j < MSLOTS; ++j) if (lane == j) val = e[j];
    ws[OFF_P + row * MSLOTS + lane] = val * inv;
  }
}

// ------------- image pass: norms, s, argmax, column max (wave per row) -------------
__global__ void k_img_stats(const float* __restrict__ image, const float* __restrict__ cache,
                            float* __restrict__ ws) {
  __shared__ unsigned cmk[32];
  const int tid = threadIdx.x, lane = tid & 31;
  const int row = blockIdx.x * 8 + (tid >> 5);
  if (tid < 32) cmk[tid] = 0u;
  __syncthreads();

  const float4* x4 = reinterpret_cast<const float4*>(image) + row * (DIM / 4);
  const float4* c4 = reinterpret_cast<const float4*>(cache);
  float part[MSLOTS], nrm = 0.0f;
#pragma unroll
  for (int j = 0; j < MSLOTS; ++j) part[j] = 0.0f;
  for (int k = lane; k < DIM / 4; k += 32) {
    float4 xv = x4[k];
    nrm += xv.x * xv.x + xv.y * xv.y + xv.z * xv.z + xv.w * xv.w;
#pragma unroll
    for (int j = 0; j < MSLOTS; ++j) {
      float4 cv = c4[j * (DIM / 4) + k];
      part[j] += xv.x * cv.x + xv.y * cv.y + xv.z * cv.z + xv.w * cv.w;
    }
  }
#pragma unroll
  for (int off = 16; off > 0; off >>= 1) {
    nrm += __shfl_xor(nrm, off, 32);
#pragma unroll
    for (int j = 0; j < MSLOTS; ++j) part[j] += __shfl_xor(part[j], off, 32);
  }
  float invn = 1.0f / fmaxf(sqrtf(nrm), 1e-12f);
  float s[MSLOTS];
#pragma unroll
  for (int j = 0; j < MSLOTS; ++j) s[j] = part[j] * invn;
  int a = 0; float best = s[0];
#pragma unroll
  for (int j = 1; j < MSLOTS; ++j) if (s[j] > best) { best = s[j]; a = j; }
  if (lane == 0) {
    ((int*)(ws + OFF_ASSIGN))[row] = a;
    ws[OFF_SMAX + row] = best;
    ws[OFF_INVN + row] = invn;
  }
  if (lane < MSLOTS) {
    float mys = 0.0f;
#pragma unroll
    for (int j = 0; j < MSLOTS; ++j) if (lane == j) mys = s[j];
    atomicMax(&cmk[lane], fkey(mys));
  }
  __syncthreads();
  if (tid < MSLOTS) atomicMax(&((unsigned*)(ws + OFF_CMAX))[tid], cmk[tid]);
}

// ---------------- WMMA GEMM: out = ALPHA*(text@W1^T + p@G) + text ----------------
// K = 1024 (text/W) + one padded block of 32 (p/G). Block tile 64x128, 8 waves,
// each wave: 16x64 via 4 accumulators of v_wmma_f32_16x16x4_f32.
__global__ void __launch_bounds__(256)
k_gemm(const float* __restrict__ text, const float* __restrict__ W,
       const float* __restrict__ ws, float* __restrict__ dout, float* __restrict__ ws_rn) {
  __shared__ float As[64][33];
  __shared__ float Bs[32][132];
  const int tid = threadIdx.x;
  const int lane = tid & 31, wave = tid >> 5;
  const int wm = wave >> 1, wn = wave & 1;
  const int hl = lane & 15, hh = lane >> 4;
  const int row0 = blockIdx.y * 64, col0 = blockIdx.x * 128;
  const float* p = ws + OFF_P;
  const float* G = ws + OFF_G;

  v8f acc[4] = {};

  for (int kb = 0; kb < 33; ++kb) {
    // ---- stage A tile: 64 rows x 32 k ----
#pragma unroll
    for (int it = 0; it < 2; ++it) {
      int r = (tid >> 3) + it * 32, kq = tid & 7;
      float vals[4];
      if (kb < 32) {
        const float4* t4 = reinterpret_cast<const float4*>(text);
        float4 v = t4[(row0 + r) * (DIM / 4) + kb * 8 + kq];
        vals[0] = v.x; vals[1] = v.y; vals[2] = v.z; vals[3] = v.w;
      } else {
#pragma unroll
        for (int c = 0; c < 4; ++c) {
          int j = kq * 4 + c;
          vals[c] = (j < MSLOTS) ? p[(row0 + r) * MSLOTS + j] : 0.0f;
        }
      }
#pragma unroll
      for (int c = 0; c < 4; ++c) As[r][kq * 4 + c] = vals[c];
    }
    // ---- stage B tile: 32 k x 128 n  (B[k][n] = W_ext[n][k] or G[k][n]) ----
#pragma unroll
    for (int it = 0; it < 4; ++it) {
      int idx = tid + it * 256;
      int n = idx >> 3, kq = idx & 7;
      float vals[4];
      if (kb < 32) {
        const float4* w4 = reinterpret_cast<const float4*>(W);
        float4 v = w4[(col0 + n) * 512 + kb * 8 + kq];
        if (kb + 1 < 32 && it == 0)
          __builtin_prefetch(&W[(size_t)(col0 + n) * 2048 + (kb + 1) * 32], 0, 1);
        vals[0] = v.x; vals[1] = v.y; vals[2] = v.z; vals[3] = v.w;
      } else {
#pragma unroll
        for (int c = 0; c < 4; ++c) {
          int k = kq * 4 + c;
          vals[c] = (k < MSLOTS) ? G[k * DIM + col0 + n] : 0.0f;
        }
      }
#pragma unroll
      for (int c = 0; c < 4; ++c) Bs[kq * 4 + c][n] = vals[c];
    }
    __syncthreads();
    // ---- WMMA over this K block ----
#pragma unroll
    for (int kk = 0; kk < 32; kk += 4) {
      int kb2 = kk + hh * 2;                       // lanes 16-31 hold K+2,K+3
      v2f a;
      a.x = As[wm * 16 + hl][kb2];
      a.y = As[wm * 16 + hl][kb2 + 1];
#pragma unroll
      for (int s = 0; s < 4; ++s) {
        int n = wn * 64 + s * 16 + hl;
        v2f b;
        b.x = Bs[kb2][n];
        b.y = Bs[kb2 + 1][n];
        acc[s] = __builtin_amdgcn_wmma_f32_16x16x4_f32(
            false, a, false, b, (short)0, acc[s], false, false);
      }
    }
    __syncthreads();
  }

  // ---- epilogue: residual, store, per-row sum of squares ----
  float sq[8];
#pragma unroll
  for (int v = 0; v < 8; ++v) sq[v] = 0.0f;
#pragma unroll
  for (int s = 0; s < 4; ++s) {
    int n = col0 + wn * 64 + s * 16 + hl;
#pragma unroll
    for (int v = 0; v < 8; ++v) {
      int r = row0 + wm * 16 + hh * 8 + v;         // C/D layout: VGPR v -> row v/v+8
      float o = ALPHA_ * acc[s][v] + text[r * DIM + n];
      dout[(size_t)r * DIM + n] = o;
      sq[v] += o * o;
    }
  }
#pragma unroll
  for (int off = 1; off <= 8; off <<= 1) {
#pragma unroll
    for (int v = 0; v < 8; ++v) sq[v] += __shfl_xor(sq[v], off, 32);
  }
  if (hl == 0) {
#pragma unroll
    for (int v = 0; v < 8; ++v)
      atomicAdd(&ws_rn[OFF_RN - OFF_RN + row0 + wm * 16 + hh * 8 + v], sq[v]);
  }
}

// ------------- scatter pass A: per-block (256 rows) partial slot sums -------------
__global__ void k_scatter_a(const float* __restrict__ image, float* __restrict__ ws) {
  __shared__ int a_s[256];
  __shared__ float coef_s[256];
  __shared__ int cnt_s[MSLOTS];
  const int tid = threadIdx.x, b = blockIdx.x;
  const int i = b * 256 + tid;
  const int* assign = (const int*)(ws + OFF_ASSIGN);
  const unsigned* cmax = (const unsigned*)(ws + OFF_CMAX);

  int a = assign[i];
  float w = expf(ws[OFF_SMAX + i] - fdec(cmax[a]));   // sq[i,a]/colmax[a]
  a_s[tid] = a;
  coef_s[tid] = w * ws[OFF_INVN + i];
  if (tid < MSLOTS) cnt_s[tid] = 0;
  __syncthreads();
  if (tid == 0) {
    for (int r = 0; r < 256; ++r) cnt_s[a_s[r]]++;
  }
  __syncthreads();
  if (tid < MSLOTS) atomicAdd(&((int*)(ws + OFF_CNT))[tid], cnt_s[tid]);

  const float4* x4 = reinterpret_cast<const float4*>(image);
  float4* part4 = reinterpret_cast<float4*>(ws + OFF_PART) + b * (MSLOTS * DIM / 4);
#pragma unroll 1
  for (int j = 0; j < MSLOTS; ++j) {
    float ax = 0.0f, ay = 0.0f, az = 0.0f, aw = 0.0f;
    for (int r = 0; r < 256; ++r) {
      if (a_s[r] == j) {                           // uniform branch across block
        float4 xv = x4[(size_t)(b * 256 + r) * (DIM / 4) + tid];
        float c = coef_s[r];
        ax += c * xv.x; ay += c * xv.y; az += c * xv.z; aw += c * xv.w;
      }
    }
    float4 o; o.x = ax; o.y = ay; o.z = az; o.w = aw;
    part4[j * (DIM / 4) + tid] = o;                // thread owns k-slice: race-free
  }
}

// ------------- scatter pass B: reduce 128 partials -> slot_sum -------------
__global__ void k_scatter_b(float* __restrict__ ws) {
  int o = blockIdx.x * 256 + threadIdx.x;          // 25600
  float s = 0.0f;
  for (int b = 0; b < 128; ++b) s += ws[OFF_PART + (size_t)b * (MSLOTS * DIM) + o];
  ws[OFF_SLOT + o] = s;
}

// ------------- cache momentum update + row L2 normalize -------------
__global__ void k_cache_upd(const float* __restrict__ cache, const float* __restrict__ ws,
                            float* __restrict__ out_cache) {
  __shared__ float upd[DIM];
  __shared__ float red[256];
  const int j = blockIdx.x, tid = threadIdx.x;
  bool has = ((const int*)(ws + OFF_CNT))[j] > 0;
  float s = 0.0f;
  for (int k = tid; k < DIM; k += 256) {
    float c = cache[j * DIM + k];
    float u = has ? (MOM_ * c + (1.0f - MOM_) * ws[OFF_SLOT + j * DIM + k]) : c;
    upd[k] = u; s += u * u;
  }
  red[tid] = s; __syncthreads();
  for (int st = 128; st > 0; st >>= 1) {
    if (tid < st) red[tid] += red[tid + st];
    __syncthreads();
  }
  float inv = 1.0f / fmaxf(sqrtf(red[0]), 1e-12f);
  for (int k = tid; k < DIM; k += 256) out_cache[j * DIM + k] = upd[k] * inv;
}

// ------------- rownorm2 -> 1/max(||row||, eps) -------------
__global__ void k_rn_inv(float* __restrict__ ws) {
  int i = blockIdx.x * 256 + threadIdx.x;
  ws[OFF_RN + i] = 1.0f / fmaxf(sqrtf(ws[OFF_RN + i]), 1e-12f);
}

// ------------- loss = mean |l2norm(out) - text| -------------
__global__ void k_loss(const float* __restrict__ dout, const float* __restrict__ text,
                       float* __restrict__ ws) {
  __shared__ float red[256];
  const int tid = threadIdx.x;
  float part = 0.0f;
  for (size_t idx = (size_t)blockIdx.x * 256 + tid; idx < (size_t)C_ROWS * DIM;
       idx += (size_t)gridDim.x * 256) {
    size_t row = idx >> 10;
    part += fabsf(dout[idx] * ws[OFF_RN + row] - text[idx]);
  }
  red[tid] = part; __syncthreads();
  for (int st = 128; st > 0; st >>= 1) {
    if (tid < st) red[tid] += red[tid + st];
    __syncthreads();
  }
  if (tid == 0) atomicAdd(ws + OFF_LOSS, red[0]);
}

__global__ void k_finalize(const float* __restrict__ ws, float* __restrict__ dout) {
  dout[(size_t)C_ROWS * DIM] = ws[OFF_LOSS] * (1.0f / ((float)C_ROWS * (float)DIM));
}

// ---------------------------------------------------------------------------
extern "C" void kernel_launch(void* const* d_in, const int* in_sizes, int n_in,
                              void* d_out, int out_size, void* d_ws, size_t ws_size,
                              hipStream_t stream) {
  (void)in_sizes; (void)n_in; (void)out_size; (void)ws_size;
  const float* text  = (const float*)d_in[0];
  const float* image = (const float*)d_in[1];
  const float* cache = (const float*)d_in[2];
  const float* W     = (const float*)d_in[3];
  float* out = (float*)d_out;
  float* ws  = (float*)d_ws;                       // needs ~16.4 MB

  k_init<<<dim3(129), dim3(256), 0, stream>>>(ws);
  k_G<<<dim3(100), dim3(256), 0, stream>>>(cache, W, ws);
  k_softmax_p<<<dim3(C_ROWS / 8), dim3(256), 0, stream>>>(text, cache, ws);
  k_img_stats<<<dim3(C_ROWS / 8), dim3(256), 0, stream>>>(image, cache, ws);
  k_gemm<<<dim3(DIM / 128, C_ROWS / 64), dim3(256), 0, stream>>>(text, W, ws, out,
                                                                 ws + OFF_RN);
  k_scatter_a<<<dim3(C_ROWS / 256), dim3(256), 0, stream>>>(image, ws);
  k_scatter_b<<<dim3(100), dim3(256), 0, stream>>>(ws);
  k_cache_upd<<<dim3(MSLOTS), dim3(256), 0, stream>>>(cache, ws,
                                                      out + (size_t)C_ROWS * DIM + 1);
  k_rn_inv<<<dim3(C_ROWS / 256), dim3(256), 0, stream>>>(ws);
  k_loss<<<dim3(2048), dim3(256), 0, stream>>>(out, text, ws);
  k_finalize<<<dim3(1), dim3(1), 0, stream>>>(ws, out);
}